// ParameterTypeEmbedding_5446018531779
// MI455X (gfx1250) — compile-verified
//
#include <hip/hip_runtime.h>

typedef float v2f __attribute__((ext_vector_type(2)));
typedef float v8f __attribute__((ext_vector_type(8)));

#define ROWS_TOTAL (4 * 4096)   // B*N
#define FDIM 64
#define EDIM 64

// out[row, f*64+e] = xn[row,f]*W[f,e] + (pb[f,e]+pe[f,e]),
// computed as D = diag(xn) x W + C with V_WMMA_F32_16X16X4_F32.
__global__ __launch_bounds__(512) void pte_wmma_kernel(
    const float* __restrict__ x,
    const float* __restrict__ gamma,
    const float* __restrict__ beta,
    const float* __restrict__ w,
    const float* __restrict__ pb,
    const float* __restrict__ pe,
    float* __restrict__ out)
{
    const int lane = threadIdx.x & 31;
    const int wave = threadIdx.x >> 5;   // 0..15: one 16x16 output tile each
    const int fi   = wave >> 2;          // f-tile 0..3
    const int ei   = wave & 3;           // e-tile 0..3
    const int m    = lane & 15;          // column (N) index within tile
    const int hi   = lane >> 4;          // lane half

    // ---- row-invariant operands, held in registers for the whole loop ----
    // B slices for K-chunks j=0..3 of this tile's diagonal block.
    // B 4x16 f32 layout: VGPR0 = {K=0 (lanes 0-15), K=2 (lanes 16-31)},
    //                    VGPR1 = {K=1,              K=3}.
    v2f bm[4];
#pragma unroll
    for (int j = 0; j < 4; ++j) {
        const int f0 = fi * 16 + j * 4 + 2 * hi;
        bm[j][0] = w[(size_t)f0 * EDIM + ei * 16 + m];
        bm[j][1] = w[(size_t)(f0 + 1) * EDIM + ei * 16 + m];
    }
    // C tile (16x16 f32): VGPR r = row M = r + 8*hi, N = m.
    v8f cmat;
#pragma unroll
    for (int r = 0; r < 8; ++r) {
        const int f   = fi * 16 + r + 8 * hi;
        const int idx = f * EDIM + ei * 16 + m;
        cmat[r] = pb[idx] + pe[idx];
    }
    // LayerNorm affine params for this lane's two features.
    const float g0  = gamma[2 * lane],  g1  = gamma[2 * lane + 1];
    const float be0 = beta[2 * lane],   be1 = beta[2 * lane + 1];

    for (int row = blockIdx.x; row < ROWS_TOTAL; row += gridDim.x) {
        // Each lane loads 2 of the 64 features (full row -> wave reduction).
        const float2 xv = *(const float2*)(x + (size_t)row * FDIM + 2 * lane);
        float s  = xv.x + xv.y;
        float ss = xv.x * xv.x + xv.y * xv.y;
#pragma unroll
        for (int o = 16; o >= 1; o >>= 1) {
            s  += __shfl_xor(s,  o, 32);
            ss += __shfl_xor(ss, o, 32);
        }
        const float mean = s * (1.0f / 64.0f);
        const float var  = ss * (1.0f / 64.0f) - mean * mean;
        const float rstd = __frsqrt_rn(var + 1e-5f);
        const float xn0  = (xv.x - mean) * rstd * g0 + be0;
        const float xn1  = (xv.y - mean) * rstd * g1 + be1;

        // Broadcast xn[fi*16 + m] to this lane (holder lane = g>>1, comp g&1).
        const int   src = fi * 8 + (m >> 1);
        const float t0  = __shfl(xn0, src, 32);
        const float t1  = __shfl(xn1, src, 32);
        const float xnv = (m & 1) ? t1 : t0;

        // D = diag(xn) * W + C : 4 chained K=4 fp32 WMMAs.
        v8f acc = cmat;
#pragma unroll
        for (int j = 0; j < 4; ++j) {
            // A 16x4 f32 one-hot slice: lanes 0-15 hold K=0/1 in V0/V1,
            // lanes 16-31 hold K=2/3; row M = m; nonzero iff M == 4j+Klocal.
            v2f a;
            a[0] = (m == j * 4 + 2 * hi)     ? xnv : 0.0f;
            a[1] = (m == j * 4 + 1 + 2 * hi) ? xnv : 0.0f;
            acc = __builtin_amdgcn_wmma_f32_16x16x4_f32(
                false, a, false, bm[j], (short)0, acc, false, false);
        }

        // D tile: VGPR r -> f = fi*16 + r + 8*hi, e = ei*16 + m.
        // Output (256MB) is write-once/never-reread and exceeds the 192MB L2:
        // use non-temporal stores (TH=NT) to avoid L2 pollution.
        float* o = out + (size_t)row * (FDIM * EDIM)
                       + (size_t)(fi * 16 + 8 * hi) * EDIM + ei * 16 + m;
#pragma unroll
        for (int r = 0; r < 8; ++r)
            __builtin_nontemporal_store(acc[r], o + (size_t)r * EDIM);
    }
}

extern "C" void kernel_launch(void* const* d_in, const int* in_sizes, int n_in,
                              void* d_out, int out_size, void* d_ws, size_t ws_size,
                              hipStream_t stream) {
    const float* x  = (const float*)d_in[0];
    const float* g  = (const float*)d_in[1];
    const float* b  = (const float*)d_in[2];
    const float* w  = (const float*)d_in[3];
    const float* pb = (const float*)d_in[4];
    const float* pe = (const float*)d_in[5];
    float* out = (float*)d_out;

    // 2048 blocks x 512 threads (16 waves each); each block grid-strides
    // over 8 of the 16384 rows.
    pte_wmma_kernel<<<dim3(2048), dim3(512), 0, stream>>>(x, g, b, w, pb, pe, out);
}